// Model_47536698032993
// MI455X (gfx1250) — compile-verified
//
#include <hip/hip_runtime.h>
#include <math.h>

// ---- problem constants -----------------------------------------------------
#define D_MODEL 256
#define N_HEADS 8
#define HEAD    32
#define HW      2500      // 50*50 tokens after padded 1x1 conv
#define HPAD    50
#define NT      157       // ceil(2500/16) 16-wide tiles
#define GROW    768       // G row stride (3*D_MODEL)
#define GPAD    2512      // token rows padded to NT*16
#define P_DROP  0.1f
#define DROP_SCALE (1.0f/0.9f)

typedef __attribute__((ext_vector_type(2))) float v2f;
typedef __attribute__((ext_vector_type(8))) float v8f;

// D = A(16x4 f32) * B(4x16 f32) + C(16x16 f32), full fp32 precision.
__device__ __forceinline__ v8f wmma4(v2f a, v2f b, v8f c) {
  return __builtin_amdgcn_wmma_f32_16x16x4_f32(false, a, false, b,
                                               (short)0, c, false, false);
}

// ---- pass A: G[token][o] = sum_c xpad[c][token] * qkv_w[o][c] + qkv_b[o] ---
// A = x_pad^T (M=tokens, K=channels), B = qkv_w^T (K=channels, N=out-ch).
// Padding handled branch-free: clamp address, multiply by 0/1 mask.
__global__ __launch_bounds__(32)
void qkv_gemm(const float* __restrict__ x,      // (256,48,48)
              const float* __restrict__ w,      // (768,256)
              const float* __restrict__ b,      // (768)
              float* __restrict__ G) {          // (GPAD,768)
  const int lane = threadIdx.x;
  const int half = lane >> 4;
  const int lid  = lane & 15;
  const int t0 = blockIdx.x * 16;
  const int o0 = blockIdx.y * 16;

  const int tok = t0 + lid;                 // <= 2511
  const int hp = tok / HPAD, wp = tok % HPAD;
  const bool tvalid = (hp >= 1) & (hp <= 48) & (wp >= 1) & (wp <= 48);
  const float msk = tvalid ? 1.0f : 0.0f;   // branch-free pad mask
  const int   xb  = tvalid ? (hp - 1) * 48 + (wp - 1) : 0;  // always in-bounds
  const float* xp   = x + (size_t)(2 * half) * 2304 + xb;
  const float* wrow = w + (size_t)(o0 + lid) * D_MODEL + 2 * half;

  v8f acc = {};
  for (int kk = 0; kk < D_MODEL; kk += 4) {
    const float* xc = xp + (size_t)kk * 2304;
    v2f a;
    a.x = xc[0]    * msk;                   // unconditional loads
    a.y = xc[2304] * msk;
    v2f bb = *(const v2f*)(wrow + kk);
    acc = wmma4(a, bb, acc);
  }
  const float bias = b[o0 + lid];
  float* g = G + (size_t)(t0 + 8 * half) * GROW + o0 + lid;
#pragma unroll
  for (int r = 0; r < 8; ++r)
    g[(size_t)r * GROW] = acc[r] + bias;
}

// ---- scatter q, v views out of G ------------------------------------------
__global__ void qv_copy(const float* __restrict__ G,
                        float* __restrict__ qout, float* __restrict__ vout) {
  int i = blockIdx.x * blockDim.x + threadIdx.x;
  const int n = N_HEADS * HW * HEAD;        // 640000
  if (i >= n) return;
  const int h = i / (HW * HEAD);
  const int t = (i / HEAD) % HW;
  const int d = i % HEAD;
  const float* g = G + (size_t)t * GROW + h * HEAD + d;
  qout[i] = g[0];                           // Q channels [0,256)
  vout[i] = g[2 * D_MODEL];                 // V channels [512,768)
}

// ---- pass B: scaled_qk[h][m][n] = 4 * sum_d Q[h][m][d] * K[h][n][d] -------
__global__ __launch_bounds__(32)
void scores(const float* __restrict__ G, float* __restrict__ Sq) {
  const int lane = threadIdx.x, half = lane >> 4, lid = lane & 15;
  const int n0 = blockIdx.x * 16, m0 = blockIdx.y * 16, h = blockIdx.z;
  const float* qrow = G + (size_t)(m0 + lid) * GROW + h * HEAD + 2 * half;
  const float* krow = G + (size_t)(n0 + lid) * GROW + D_MODEL + h * HEAD + 2 * half;
  v8f acc = {};
#pragma unroll
  for (int kk = 0; kk < HEAD; kk += 4) {
    v2f a  = *(const v2f*)(qrow + kk);
    v2f bb = *(const v2f*)(krow + kk);
    acc = wmma4(a, bb, acc);
  }
  float* out = Sq + (size_t)h * HW * HW;
  const int col = n0 + lid;
  if (col < HW) {
#pragma unroll
    for (int r = 0; r < 8; ++r) {
      const int row = m0 + r + 8 * half;
      if (row < HW) out[(size_t)row * HW + col] = acc[r] * 4.0f; // *256^0.25
    }
  }
}

// ---- pass C1: per-row online (max, sum of exp) ----------------------------
__global__ __launch_bounds__(256)
void rowstats(const float* __restrict__ Sq,
              float* __restrict__ rmax, float* __restrict__ rsuminv) {
  const int row  = blockIdx.x * 8 + (threadIdx.x >> 5);
  const int lane = threadIdx.x & 31;
  if (row >= N_HEADS * HW) return;
  const float* p = Sq + (size_t)row * HW;
  float m = -3.402823466e38f, s = 0.0f;
  for (int j = lane; j < HW; j += 32) {
    const float x = p[j];
    const float mn = fmaxf(m, x);
    s = s * expf(m - mn) + expf(x - mn);
    m = mn;
  }
#pragma unroll
  for (int off = 16; off > 0; off >>= 1) {
    const float m2 = __shfl_xor(m, off, 32);
    const float s2 = __shfl_xor(s, off, 32);
    const float mn = fmaxf(m, m2);
    s = s * expf(m - mn) + s2 * expf(m2 - mn);
    m = mn;
  }
  if (lane == 0) { rmax[row] = m; rsuminv[row] = 1.0f / s; }
}

// ---- pass C2: softmax + dropout, float4 vectorized (2500 % 4 == 0) --------
__global__ void softmax_drop(const float* __restrict__ Sq,
                             const float* __restrict__ mask,
                             const float* __restrict__ rmax,
                             const float* __restrict__ rsuminv,
                             float* __restrict__ SM, float* __restrict__ DQ) {
  const size_t nvec   = (size_t)N_HEADS * HW * (HW / 4);  // 12.5M float4
  const size_t stride = (size_t)gridDim.x * blockDim.x;
  for (size_t i = blockIdx.x * (size_t)blockDim.x + threadIdx.x;
       i < nvec; i += stride) {
    const size_t row = i / (HW / 4);
    const float m  = rmax[row];
    const float rs = rsuminv[row];
    const float4 s = ((const float4*)Sq)[i];
    float4 sm;
    sm.x = expf(s.x - m) * rs;
    sm.y = expf(s.y - m) * rs;
    sm.z = expf(s.z - m) * rs;
    sm.w = expf(s.w - m) * rs;
    ((float4*)SM)[i] = sm;
    const float4 mk = ((const float4*)mask)[i];
    float4 d;
    d.x = (mk.x > P_DROP) ? sm.x * DROP_SCALE : 0.0f;
    d.y = (mk.y > P_DROP) ? sm.y * DROP_SCALE : 0.0f;
    d.z = (mk.z > P_DROP) ? sm.z * DROP_SCALE : 0.0f;
    d.w = (mk.w > P_DROP) ? sm.w * DROP_SCALE : 0.0f;
    ((float4*)DQ)[i] = d;
  }
}

// ---- pass D: AO[m][h*32+d] = sum_k P[h][m][k] * V[h][k][d] ----------------
// K dim = 2500 keys = 625 exact steps of 4; two 16-wide N accumulators.
__global__ __launch_bounds__(32)
void attn_v(const float* __restrict__ DQ, const float* __restrict__ G,
            float* __restrict__ AO) {
  const int lane = threadIdx.x, half = lane >> 4, lid = lane & 15;
  const int m0 = blockIdx.x * 16, h = blockIdx.y;
  int arow = m0 + lid; if (arow > HW - 1) arow = HW - 1;   // clamp pad rows
  const float* prow = DQ + (size_t)h * HW * HW + (size_t)arow * HW + 2 * half;
  const float* vb   = G + 2 * D_MODEL + h * HEAD + lid;    // V col base
  v8f acc0 = {}, acc1 = {};
  for (int kk = 0; kk < HW; kk += 4) {
    const v2f a = *(const v2f*)(prow + kk);
    const float* vr = vb + (size_t)(kk + 2 * half) * GROW;
    v2f b0, b1;
    b0.x = vr[0];   b0.y = vr[GROW];
    b1.x = vr[16];  b1.y = vr[GROW + 16];
    acc0 = wmma4(a, b0, acc0);
    acc1 = wmma4(a, b1, acc1);
  }
  float* ao = AO + (size_t)(m0 + 8 * half) * D_MODEL + h * HEAD + lid;
#pragma unroll
  for (int r = 0; r < 8; ++r) {
    ao[(size_t)r * D_MODEL]      = acc0[r];
    ao[(size_t)r * D_MODEL + 16] = acc1[r];
  }
}

// ---- pass E: out[o][t] = sum_c out_w[o][c] * AO[t][c] + out_b[o] ----------
__global__ __launch_bounds__(32)
void out_proj(const float* __restrict__ AO, const float* __restrict__ w,
              const float* __restrict__ b, float* __restrict__ out) {
  const int lane = threadIdx.x, half = lane >> 4, lid = lane & 15;
  const int t0 = blockIdx.x * 16, o0 = blockIdx.y * 16;
  const float* wr = w  + (size_t)(o0 + lid) * D_MODEL + 2 * half;
  const float* ar = AO + (size_t)(t0 + lid) * D_MODEL + 2 * half;
  v8f acc = {};
  for (int kk = 0; kk < D_MODEL; kk += 4) {
    v2f a  = *(const v2f*)(wr + kk);
    v2f bb = *(const v2f*)(ar + kk);
    acc = wmma4(a, bb, acc);
  }
  const int col = t0 + lid;
  if (col < HW) {
#pragma unroll
    for (int r = 0; r < 8; ++r) {
      const int o = o0 + r + 8 * half;
      out[(size_t)o * HW + col] = acc[r] + b[o];
    }
  }
}

// ---- launch ---------------------------------------------------------------
extern "C" void kernel_launch(void* const* d_in, const int* in_sizes, int n_in,
                              void* d_out, int out_size, void* d_ws, size_t ws_size,
                              hipStream_t stream) {
  const float* x1    = (const float*)d_in[0];
  const float* qkv_w = (const float*)d_in[3];
  const float* qkv_b = (const float*)d_in[4];
  const float* out_w = (const float*)d_in[5];
  const float* out_b = (const float*)d_in[6];
  const float* dmask = (const float*)d_in[7];

  float* out   = (float*)d_out;              // (1,256,50,50)   640000
  float* q_out = out   + 640000;             // (8,2500,32)     640000
  float* v_out = q_out + 640000;             // (8,2500,32)     640000
  float* SM    = v_out + 640000;             // softmax_qk   50000000
  float* DQ    = SM    + 50000000;           // dropout_qk   50000000
  float* Sq    = DQ    + 50000000;           // scaled_qk    50000000

  float* G       = (float*)d_ws;                         // GPAD*768
  float* AO      = G  + (size_t)GPAD * GROW;             // GPAD*256
  float* rmax    = AO + (size_t)GPAD * D_MODEL;          // 8*2500
  float* rsuminv = rmax + N_HEADS * HW;                  // 8*2500

  qkv_gemm   <<<dim3(NT, 48),          32, 0, stream>>>(x1, qkv_w, qkv_b, G);
  qv_copy    <<<(640000 + 255) / 256, 256, 0, stream>>>(G, q_out, v_out);
  scores     <<<dim3(NT, NT, N_HEADS), 32, 0, stream>>>(G, Sq);
  rowstats   <<<(N_HEADS * HW + 7) / 8, 256, 0, stream>>>(Sq, rmax, rsuminv);
  softmax_drop<<<4096,                 256, 0, stream>>>(Sq, dmask, rmax, rsuminv, SM, DQ);
  attn_v     <<<dim3(NT, N_HEADS),     32, 0, stream>>>(DQ, G, AO);
  out_proj   <<<dim3(NT, 16),          32, 0, stream>>>(AO, out_w, out_b, out);
}